// SinkhornDistance_Mahal_30021821399472
// MI455X (gfx1250) — compile-verified
//
#include <hip/hip_runtime.h>
#include <math.h>

// Problem constants (from reference)
#define NB 4
#define PP1 2048
#define PP2 2048
#define DD 64
#define EPS 0.1f
#define MAX_ITER 50
#define THRESH 0.1f

typedef float v2f __attribute__((ext_vector_type(2)));
typedef float v8f __attribute__((ext_vector_type(8)));

// Workspace layout (float indices). Total ~704513 floats (~2.8 MB).
#define WS_ASYM   0                         // 64*64
#define WS_XA     (WS_ASYM + DD*DD)         // NB*P1*DD = 524288
#define WS_DX     (WS_XA + NB*PP1*DD)       // NB*P1
#define WS_DY     (WS_DX + NB*PP1)          // NB*P2
#define WS_U      (WS_DY + NB*PP2)          // NB*P1
#define WS_V      (WS_U + NB*PP1)           // NB*P2
#define WS_DU     (WS_V + NB*PP2)           // NB*P1
#define WS_PM     (WS_DU + NB*PP1)          // NB*8*P2
#define WS_PS     (WS_PM + NB*8*PP2)        // NB*8*P2
#define WS_CP     (WS_PS + NB*8*PP2)        // 4096 cost partials
#define WS_ACTIVE (WS_CP + 4096)            // 1 int

// ---------------- setup: Asym = M + M^T, u=v=0, active=1 ----------------
__global__ void k_sym_init(const float* __restrict__ M, float* __restrict__ ws) {
  int g = blockIdx.x * 256 + threadIdx.x;            // grid covers 8192
  if (g < DD * DD) {
    int d = g >> 6, e = g & 63;
    ws[WS_ASYM + g] = M[d * DD + e] + M[e * DD + d];
  }
  if (g < NB * PP1) { ws[WS_U + g] = 0.0f; ws[WS_V + g] = 0.0f; }
  if (g == 0) ((int*)ws)[WS_ACTIVE] = 1;
}

// ---------------- xA[n,i,:] = x[n,i,:] @ Asym ----------------
__global__ void k_xa(const float* __restrict__ x, float* __restrict__ ws) {
  int g = blockIdx.x * 256 + threadIdx.x;            // < NB*P1*DD
  int e = g & 63;
  int row = g >> 6;
  const float* xr = x + (size_t)row * DD;
  const float* A = ws + WS_ASYM;
  float acc = 0.0f;
#pragma unroll 8
  for (int d = 0; d < DD; ++d) acc = fmaf(xr[d], A[d * DD + e], acc);
  ws[WS_XA + g] = acc;
}

// ---------------- dx_i = 0.5*xA_i.x_i ; dy_j = 0.5*(y_j A).y_j ----------------
__global__ void k_dxdy(const float* __restrict__ x, const float* __restrict__ y,
                       float* __restrict__ ws) {
  int g = blockIdx.x * 256 + threadIdx.x;            // grid covers 16384
  const float* A = ws + WS_ASYM;
  if (g < NB * PP1) {
    const float* xr = x + (size_t)g * DD;
    const float* xa = ws + WS_XA + (size_t)g * DD;
    float acc = 0.0f;
#pragma unroll 8
    for (int d = 0; d < DD; ++d) acc = fmaf(xa[d], xr[d], acc);
    ws[WS_DX + g] = 0.5f * acc;
  } else if (g < 2 * NB * PP2) {
    int r = g - NB * PP1;
    const float* yr = y + (size_t)r * DD;
    float acc = 0.0f;
    for (int d = 0; d < DD; ++d) {
      float t = 0.0f;
#pragma unroll 8
      for (int e = 0; e < DD; ++e) t = fmaf(A[d * DD + e], yr[e], t);
      acc = fmaf(yr[d], t, acc);
    }
    ws[WS_DY + r] = 0.5f * acc;
  }
}

// ---------------- C tile via fp32 WMMA: C = dx + dy - xA @ y^T ----------------
// One 16x16 tile per wave; 8 waves per 256-thread block; 16 K-steps of K=4.
__global__ void k_cost(const float* __restrict__ y, const float* __restrict__ ws,
                       float* __restrict__ Cout) {
  int wave = threadIdx.x >> 5;
  int lane = threadIdx.x & 31;
  int tile = blockIdx.x * 8 + wave;                  // 65536 tiles total
  int n  = tile >> 14;
  int it = (tile >> 7) & 127;
  int jt = tile & 127;
  int i0 = it * 16, j0 = jt * 16;
  int half = lane >> 4;                              // K-pair select
  int l = lane & 15;                                 // row (A) / col (B)

  const float2* xa2 = (const float2*)(ws + WS_XA + ((size_t)(n * PP1 + i0 + l)) * DD);
  const float2* y2  = (const float2*)(y + ((size_t)(n * PP2 + j0 + l)) * DD);

  v8f acc = {0.f, 0.f, 0.f, 0.f, 0.f, 0.f, 0.f, 0.f};
#pragma unroll
  for (int kk = 0; kk < DD; kk += 4) {
    // A 16x4 layout: v0 = K (0 or 2), v1 = K+1; half selects K base 0 vs 2
    float2 av = xa2[(kk >> 1) + half];
    float2 bv = y2[(kk >> 1) + half];
    v2f a = {av.x, av.y};
    v2f b = {bv.x, bv.y};
    acc = __builtin_amdgcn_wmma_f32_16x16x4_f32(false, a, false, b,
                                                (short)0, acc, false, false);
  }

  float dyv = ws[WS_DY + n * PP2 + j0 + l];
  size_t base = (size_t)n * PP1 * PP2;
#pragma unroll
  for (int r = 0; r < 8; ++r) {
    int i = i0 + r + 8 * half;                       // D layout: M = r (+8 for hi lanes)
    float c = ws[WS_DX + n * PP1 + i] + dyv - acc[r];
    Cout[base + (size_t)i * PP2 + (j0 + l)] = c;
  }
}

// online (max,sumexp) merge helper
__device__ __forceinline__ void lse_merge(float& m, float& s, float m2, float s2) {
  float mm = fmaxf(m, m2);
  s = s * expf(m - mm) + s2 * expf(m2 - mm);
  m = mm;
}

// ---------------- u update: per-row logsumexp over j ----------------
__global__ void k_rowlse(const float* __restrict__ Cmat, float* __restrict__ ws) {
  if (!((const int*)ws)[WS_ACTIVE]) return;
  int row = blockIdx.x;                              // n*P1 + i
  int n = row >> 11;
  int tid = threadIdx.x;
  float ui = ws[WS_U + row];
  const float* Crow = Cmat + (size_t)row * PP2;
  const float* v = ws + WS_V + n * PP2;
  float m = -INFINITY, s = 0.0f;
  for (int t = 0; t < PP2 / 256; ++t) {
    int j = t * 256 + tid;
    float val = (ui + v[j] - Crow[j]) / EPS;
    if (val > m) { s = s * expf(m - val) + 1.0f; m = val; }
    else         { s += expf(val - m); }
  }
  __shared__ float sm[256], ss[256];
  sm[tid] = m; ss[tid] = s; __syncthreads();
  for (int st = 128; st > 0; st >>= 1) {
    if (tid < st) {
      float mm = sm[tid], sv = ss[tid];
      lse_merge(mm, sv, sm[tid + st], ss[tid + st]);
      sm[tid] = mm; ss[tid] = sv;
    }
    __syncthreads();
  }
  if (tid == 0) {
    float lse = sm[0] + logf(ss[0]);
    float log_mu = logf(1.0f / PP1 + 1e-8f);
    float un = EPS * (log_mu - lse) + ui;
    ws[WS_U + row] = un;
    ws[WS_DU + row] = fabsf(un - ui);
  }
}

// ---------------- v update stage 1: column partial LSE over 256-row chunks ----
__global__ void k_colpart(const float* __restrict__ Cmat, float* __restrict__ ws) {
  if (!((const int*)ws)[WS_ACTIVE]) return;
  int bx = blockIdx.x;                               // 0..255 : (n, jt, chunk)
  int n = bx >> 6;
  int jt = (bx >> 3) & 7;
  int chunk = bx & 7;
  int j = jt * 256 + threadIdx.x;
  float vj = ws[WS_V + n * PP2 + j];
  const float* u = ws + WS_U + n * PP1 + chunk * 256;
  const float* Cb = Cmat + ((size_t)n * PP1 + chunk * 256) * PP2 + j;
  float m = -INFINITY, s = 0.0f;
  for (int i = 0; i < 256; ++i) {
    float val = (u[i] + vj - Cb[(size_t)i * PP2]) / EPS;
    if (val > m) { s = s * expf(m - val) + 1.0f; m = val; }
    else         { s += expf(val - m); }
  }
  ws[WS_PM + (n * 8 + chunk) * PP2 + j] = m;
  ws[WS_PS + (n * 8 + chunk) * PP2 + j] = s;
}

// ---------------- v update stage 2: merge partials, write v ----------------
__global__ void k_colmerge(float* __restrict__ ws) {
  if (!((const int*)ws)[WS_ACTIVE]) return;
  int bx = blockIdx.x;                               // 0..31 : n*8 + jt
  int n = bx >> 3;
  int jt = bx & 7;
  int j = jt * 256 + threadIdx.x;
  float m = ws[WS_PM + (n * 8) * PP2 + j];
  float s = ws[WS_PS + (n * 8) * PP2 + j];
  for (int c = 1; c < 8; ++c)
    lse_merge(m, s, ws[WS_PM + (n * 8 + c) * PP2 + j], ws[WS_PS + (n * 8 + c) * PP2 + j]);
  float lse = m + logf(s);
  float log_nu = logf(1.0f / PP2 + 1e-8f);
  float vj = ws[WS_V + n * PP2 + j];
  ws[WS_V + n * PP2 + j] = EPS * (log_nu - lse) + vj;
}

// ---------------- err = mean_n sum_i |du| ; active &= (err >= THRESH) --------
__global__ void k_active(float* __restrict__ ws) {
  __shared__ float sh[256];
  int tid = threadIdx.x;
  float acc = 0.0f;
  for (int k = tid; k < NB * PP1; k += 256) acc += ws[WS_DU + k];
  sh[tid] = acc; __syncthreads();
  for (int st = 128; st > 0; st >>= 1) {
    if (tid < st) sh[tid] += sh[tid + st];
    __syncthreads();
  }
  if (tid == 0) {
    float err = sh[0] / (float)NB;
    int* act = (int*)ws + WS_ACTIVE;
    *act = (*act && (err >= THRESH)) ? 1 : 0;
  }
}

// ---------------- pi = exp(bias), cost partials ----------------
__global__ void k_pi(const float* __restrict__ Cmat, const float* __restrict__ ws,
                     float* __restrict__ out) {
  int tid = threadIdx.x;
  size_t base = (size_t)blockIdx.x * 4096;           // contiguous, within one n
  int n = (int)(base >> 22);
  const float* u = ws + WS_U + n * PP1;
  const float* v = ws + WS_V + n * PP2;
  float acc = 0.0f;
#pragma unroll 4
  for (int t = 0; t < 16; ++t) {
    size_t idx = base + (size_t)t * 256 + tid;
    int rem = (int)(idx & ((size_t)PP1 * PP2 - 1));
    int i = rem >> 11, j = rem & 2047;
    float c = Cmat[idx];
    float p = expf((u[i] + v[j] - c) / EPS);
    out[4 + idx] = p;                                // pi region
    acc = fmaf(p, c, acc);
  }
  __shared__ float sh[256];
  sh[tid] = acc; __syncthreads();
  for (int st = 128; st > 0; st >>= 1) {
    if (tid < st) sh[tid] += sh[tid + st];
    __syncthreads();
  }
  if (tid == 0) ((float*)ws)[WS_CP + blockIdx.x] = sh[0];
}

__global__ void k_costred(const float* __restrict__ ws, float* __restrict__ out) {
  int n = blockIdx.x, tid = threadIdx.x;
  __shared__ float sh[256];
  float acc = 0.0f;
  for (int k = tid; k < 1024; k += 256) acc += ws[WS_CP + n * 1024 + k];
  sh[tid] = acc; __syncthreads();
  for (int st = 128; st > 0; st >>= 1) {
    if (tid < st) sh[tid] += sh[tid + st];
    __syncthreads();
  }
  if (tid == 0) out[n] = sh[0];
}

extern "C" void kernel_launch(void* const* d_in, const int* in_sizes, int n_in,
                              void* d_out, int out_size, void* d_ws, size_t ws_size,
                              hipStream_t stream) {
  const float* x = (const float*)d_in[0];   // [4,2048,64]
  const float* y = (const float*)d_in[1];   // [4,2048,64]
  const float* M = (const float*)d_in[2];   // [64,64]
  float* out = (float*)d_out;               // cost[4] | pi[4,2048,2048] | C[4,2048,2048]
  float* ws  = (float*)d_ws;
  float* Cout = out + 4 + (size_t)NB * PP1 * PP2;  // C region (pi is out+4)

  k_sym_init<<<32, 256, 0, stream>>>(M, ws);
  k_xa<<<(NB * PP1 * DD) / 256, 256, 0, stream>>>(x, ws);
  k_dxdy<<<(2 * NB * PP1) / 256, 256, 0, stream>>>(x, y, ws);
  k_cost<<<(NB * (PP1 / 16) * (PP2 / 16)) / 8, 256, 0, stream>>>(y, ws, Cout);

  for (int it = 0; it < MAX_ITER; ++it) {
    k_rowlse<<<NB * PP1, 256, 0, stream>>>(Cout, ws);
    k_colpart<<<NB * 8 * 8, 256, 0, stream>>>(Cout, ws);
    k_colmerge<<<NB * 8, 256, 0, stream>>>(ws);
    k_active<<<1, 256, 0, stream>>>(ws);
  }

  k_pi<<<((size_t)NB * PP1 * PP2) / 4096, 256, 0, stream>>>(Cout, ws, out);
  k_costred<<<NB, 256, 0, stream>>>(ws, out);
}